// RoiLayer_25134148616736
// MI455X (gfx1250) — compile-verified
//
#include <hip/hip_runtime.h>
#include <math.h>

// ROI adaptive max-pool for MI455X (gfx1250).
// Memory-bound gather/max (no matmul content -> no WMMA). Uses the CDNA5
// async-tensor data path: GLOBAL_LOAD_ASYNC_TO_LDS_B128 (ASYNCcnt) in a
// depth-4 software pipeline, DS_LOAD_B128 readback, wave32-sized blocks.

#define AS3 __attribute__((address_space(3)))
typedef __attribute__((ext_vector_type(4))) float v4f;

namespace {

constexpr int H = 64, W = 64, C = 256, P = 7;
constexpr int DEPTH = 4;              // async pipeline depth (power of 2)
constexpr int THREADS = C / 4;        // 64 threads = 2 wave32, 4 ch each

// Raw LDS-segment byte offset of a __shared__ pointer (what the async copy's
// VDST operand and ds_load's ADDR operand expect).
__device__ __forceinline__ unsigned lds_off(const void* p) {
  return (unsigned)(unsigned long long)(AS3 const char*)p;
}

// Fire-and-forget 16B copy from the global address gaddr to the LDS address lds_addr  (tracked by ASYNCcnt).
__device__ __forceinline__ void async_copy_b128(unsigned lds_addr,
                                                const float* gaddr) {
  asm volatile("global_load_async_to_lds_b128 %0, %1, off"
               :: "v"(lds_addr), "v"(gaddr)
               : "memory");
}

// ds_load_b128 with the dscnt wait fused so the result is valid on return.
__device__ __forceinline__ v4f lds_read_b128(unsigned lds_addr) {
  v4f r;
  asm volatile("ds_load_b128 %0, %1\n\t"
               "s_wait_dscnt 0"
               : "=v"(r) : "v"(lds_addr) : "memory");
  return r;
}

__global__ __launch_bounds__(THREADS) void roi_maxpool_kernel(
    const float* __restrict__ feat,   // (H, W, C)
    const float* __restrict__ rois,   // (R, 4) = x1,y1,x2,y2 normalized
    float* __restrict__ out) {        // (R, P, P, C)
  // One block per output bin (r, i, j).
  int bin = blockIdx.x;
  int j = bin % P;
  int t = bin / P;
  int i = t % P;
  int r = t / P;

  // --- bin bounds (exact replica of the reference integer math) ---
  float x1 = rois[r * 4 + 0], y1 = rois[r * 4 + 1];
  float x2 = rois[r * 4 + 2], y2 = rois[r * 4 + 3];

  int lox = (int)floorf(x1 * (float)H);
  int hix = (int)ceilf(x2 * (float)H);
  int spanx = max(hix - lox, 1);
  int xs = lox + (i * spanx) / P;                      // floor div (nonneg)
  int xe = lox + ((i + 1) * spanx + P - 1) / P;        // ceil  div (nonneg)
  int xlen = max(xe - xs, 1);
  xs = min(max(xs, 0), H - 1);

  int loy = (int)floorf(y1 * (float)W);
  int hiy = (int)ceilf(y2 * (float)W);
  int spany = max(hiy - loy, 1);
  int ys = loy + (j * spany) / P;
  int ye = loy + ((j + 1) * spany + P - 1) / P;
  int ylen = max(ye - ys, 1);
  ys = min(max(ys, 0), W - 1);

  // --- LDS staging buffers: DEPTH pixel slots + 1 trash slot ---
  __shared__ float buf[DEPTH][C];
  __shared__ float trash[C];

  int c0 = threadIdx.x * 4;           // this thread's 4 channels
  unsigned lb[DEPTH];
#pragma unroll
  for (int b = 0; b < DEPTH; ++b) lb[b] = lds_off(&buf[b][c0]);
  unsigned ltrash = lds_off(&trash[c0]);

  const int n = xlen * ylen;          // pixels in this bin (>= 1, <= ~121)

  // Issue pointer state: walks (sx, sy) over the bin region in row-major
  // order; past the end we issue trash copies so in-flight count stays DEPTH
  // and the constant-immediate s_wait_asynccnt is always correct.
  int isx = 0, isy = 0, issued = 0;
  auto issue_next = [&]() {
    if (issued < n) {
      int ix = min(xs + isx, H - 1);
      int iy = min(ys + isy, W - 1);
      const float* g = feat + (((ix * W) + iy) * C + c0);
      async_copy_b128(lb[issued & (DEPTH - 1)], g);
      if (++isy == ylen) { isy = 0; ++isx; }
    } else {
      async_copy_b128(ltrash, feat + c0);
    }
    ++issued;
  };

  // Prologue: fill the pipeline (exactly DEPTH copies in flight).
#pragma unroll
  for (int p = 0; p < DEPTH; ++p) issue_next();

  v4f acc = {-INFINITY, -INFINITY, -INFINITY, -INFINITY};

  for (int p = 0; p < n; ++p) {
    // DEPTH copies in flight; <= DEPTH-1 guarantees the oldest (slot p) is
    // complete (async load done-signals retire in issue order).
    asm volatile("s_wait_asynccnt 3" ::: "memory");
    v4f v = lds_read_b128(lb[p & (DEPTH - 1)]);
    // Slot p is fully consumed (dscnt==0 waited inside the read asm), so it
    // is safe to reissue it for pixel p+DEPTH now.
    issue_next();
    acc.x = fmaxf(acc.x, v.x);
    acc.y = fmaxf(acc.y, v.y);
    acc.z = fmaxf(acc.z, v.z);
    acc.w = fmaxf(acc.w, v.w);
  }

  // out[(r, i, j, c0..c0+3)] -- 16B aligned vector store.
  float* o = out + ((((r * P) + i) * P + j) * C + c0);
  *reinterpret_cast<v4f*>(o) = acc;
}

}  // namespace

extern "C" void kernel_launch(void* const* d_in, const int* in_sizes, int n_in,
                              void* d_out, int out_size, void* d_ws,
                              size_t ws_size, hipStream_t stream) {
  const float* feat = (const float*)d_in[0];   // (1, 64, 64, 256) fp32
  const float* rois = (const float*)d_in[1];   // (1, R, 4) fp32
  float* out = (float*)d_out;                  // (1, R, 7, 7, 256) fp32

  const int R = in_sizes[1] / 4;               // 256
  dim3 grid(R * P * P);                        // one block per output bin
  dim3 block(THREADS);                         // 64 threads = 2 wave32
  roi_maxpool_kernel<<<grid, block, 0, stream>>>(feat, rois, out);
}